// MAR_block_missingness_52441550684309
// MI455X (gfx1250) — compile-verified
//
#include <hip/hip_runtime.h>
#include <hip/hip_bf16.h>

// MAR block missingness, MI455X (gfx1250, wave32).
//
// Roofline: reading X (8192*8192*4B = 256MB) once at 23.3 TB/s ~= 11us; all
// arithmetic (~64M adds + 4096 tiny MLPs) is noise. One-pass streaming
// segmented reduction at full bandwidth (b128 vectorized, prefetch-ahead),
// LDS pre-aggregation to keep global atomics off the critical path, and the
// batched 1->3->3->1 MLP expressed as three V_WMMA_F32_16X16X4_F32 ops with
// bias folded in (ones-column in A, bias-row in B).

#define NROWS 8192
#define NCOLS 8192
#define RBK 64
#define CBK 64
#define HID 3
#define ROWS_PER_WG 8

typedef float v2f __attribute__((ext_vector_type(2)));
typedef float v4f __attribute__((ext_vector_type(4)));
typedef float v8f __attribute__((ext_vector_type(8)));

// ---------------------------------------------------------------- zero acc
__global__ void k_zero_acc(float* __restrict__ acc) {
  int i = blockIdx.x * blockDim.x + threadIdx.x;
  if (i < RBK * CBK) acc[i] = 0.0f;
}

// ------------------------------------------------- counts, cumsums, bounds
__global__ void k_counts(const int* __restrict__ row_ids,
                         const int* __restrict__ col_ids,
                         int* __restrict__ colStart,
                         int* __restrict__ rowCount,
                         int* __restrict__ colCount,
                         float* __restrict__ out) {
  __shared__ int rc[RBK];
  __shared__ int cc[CBK];
  const int t = threadIdx.x;
  if (t < RBK) rc[t] = 0;
  if (t < CBK) cc[t] = 0;
  __syncthreads();
  for (int i = t; i < NROWS; i += blockDim.x) atomicAdd(&rc[row_ids[i]], 1);
  for (int i = t; i < NCOLS; i += blockDim.x) atomicAdd(&cc[col_ids[i]], 1);
  __syncthreads();
  if (t == 0) {
    const int rowBase = RBK * CBK;            // 4096
    const int colBase = RBK * CBK + RBK + 1;  // 4161
    int r = 0, c = 0;
    out[rowBase] = 0.0f;
    out[colBase] = 0.0f;
    colStart[0] = 0;
    for (int i = 0; i < RBK; ++i) {
      rowCount[i] = rc[i];
      r += rc[i];
      out[rowBase + 1 + i] = (float)r;
    }
    for (int i = 0; i < CBK; ++i) {
      colCount[i] = cc[i];
      c += cc[i];
      colStart[i + 1] = c;
      out[colBase + 1 + i] = (float)c;
    }
  }
}

// ------------------------------------------------------- main block reduce
// Wave w of workgroup (cb, rowChunk) owns (row, colblock): the column range
// is contiguous (sorted ids). Row base is 16B aligned, so: <=3-element scalar
// head to 16B alignment, global_load_b128 bulk, scalar tail. Wave32 shuffle
// reduce, LDS pre-aggregation keyed by row-block, one global f32 atomic per
// touched (row-block, col-block) per workgroup.
__global__ void __launch_bounds__(256) k_blocksum(
    const float* __restrict__ X,
    const int* __restrict__ row_ids,
    const int* __restrict__ colStart,
    float* __restrict__ acc) {
  __shared__ float part[RBK];
  const int t = threadIdx.x;
  const int lane = t & 31;
  const int wave = t >> 5;
  const int cb = blockIdx.x;
  const int row = blockIdx.y * ROWS_PER_WG + wave;

  if (t < RBK) part[t] = 0.0f;
  __syncthreads();

  const int cs = colStart[cb];
  const int ce = colStart[cb + 1];
  const size_t base = (size_t)row * NCOLS;
  const size_t lim = (size_t)NROWS * NCOLS;

  int a0 = (cs + 3) & ~3;  // first 16B-aligned element
  if (a0 > ce) a0 = ce;
  int a1 = a0 + ((ce - a0) & ~3);  // end of aligned bulk

  float s = 0.0f;
  // head (<=3 elems)
  if (cs + lane < a0) s += X[base + cs + lane];
  // aligned bulk: float4 per lane, 128 floats per wave-iteration
  for (int c = a0 + lane * 4; c < a1; c += 128) {
    v4f v = *(const v4f*)(X + base + c);
    s += v.x + v.y + v.z + v.w;
    size_t pf = base + c + 2048;  // stream-ahead -> global_prefetch_b8
    if (pf < lim) __builtin_prefetch(&X[pf], 0, 0);
  }
  // tail (<=3 elems)
  if (a1 + lane < ce) s += X[base + a1 + lane];

  // wave32 butterfly reduce
  #pragma unroll
  for (int off = 16; off > 0; off >>= 1) s += __shfl_xor(s, off, 32);

  if (lane == 0) atomicAdd(&part[row_ids[row]], s);
  __syncthreads();
  if (t < RBK) {
    float v = part[t];
    if (v != 0.0f) atomicAdd(&acc[t * CBK + cb], v);
  }
}

// --------------------------------------------------------- MLP via WMMA
// One wave per 16 cells. Each layer is one V_WMMA_F32_16X16X4_F32:
//   A[m] = [h0,h1,h2,1], B rows = [W; bias]  =>  D = H*W + bias in one op.
// The full 16x16 ReLU'd D-tile is stashed to LDS UNCONDITIONALLY (cols >= 3
// are exactly 0 since B is zero-padded), and the ones-column of A is
// synthesized at load time with a branch-free select -- no EXEC juggling
// between WMMAs.
//
// Operand striping per ISA 7.12.2 (wave32):
//   A 16x4 : lanes 0-15 hold M=lane {v0:K0, v1:K1}; lanes 16-31 {v0:K2, v1:K3}
//   B 4x16 : lanes 0-15 hold N=lane {v0:K0, v1:K1}; lanes 16-31 {v0:K2, v1:K3}
//   D 16x16: vgpr v, lanes 0-15 -> (M=v, N=lane); lanes 16-31 -> (M=v+8)
__global__ void __launch_bounds__(32) k_mlp_wmma(
    const float* __restrict__ acc,
    const int* __restrict__ rowCount,
    const int* __restrict__ colCount,
    const float* __restrict__ W1, const float* __restrict__ b1,
    const float* __restrict__ W2, const float* __restrict__ b2,
    const float* __restrict__ W3, const float* __restrict__ b3,
    float* __restrict__ out) {
  __shared__ float h[16 * 16];
  const int lane = threadIdx.x & 31;
  const int half = lane >> 4;
  const int lm = lane & 15;
  const int cellBase = blockIdx.x * 16;
  const int cell = cellBase + lm;
  const int rb = cell >> 6;
  const int cbk = cell & 63;
  const float cnt = (float)rowCount[rb] * (float)colCount[cbk];
  const float x = acc[cell] * __builtin_amdgcn_rcpf(cnt);

  v2f a, b;

  // ---- layer 1: A[m] = [x,0,0,1], B[0][n]=W1[n], B[3][n]=b1[n]
  a[0] = half ? 0.0f : x;    // K = 2*half
  a[1] = half ? 1.0f : 0.0f; // K = 2*half + 1
  b[0] = (!half && lm < HID) ? W1[lm] : 0.0f; // row K0 (K2 row is zero)
  b[1] = ( half && lm < HID) ? b1[lm] : 0.0f; // row K3 (K1 row is zero)
  v8f d1 = {};
  d1 = __builtin_amdgcn_wmma_f32_16x16x4_f32(false, a, false, b,
                                             (short)0, d1, false, false);
  #pragma unroll
  for (int v = 0; v < 8; ++v) {  // unconditional full-tile ReLU stash
    const int m = v + 8 * half;
    h[m * 16 + lm] = d1[v] > 0.0f ? d1[v] : 0.0f;
  }
  __syncthreads();

  // ---- layer 2: B[k][n] = W2[k][n] (k<3), B[3][n] = b2[n]
  a[0] = h[lm * 16 + 2 * half];                 // cols 0 / 2
  a[1] = half ? 1.0f : h[lm * 16 + 1];          // ones col synthesized
  b[0] = (lm < HID) ? (half ? W2[2 * HID + lm] : W2[0 * HID + lm]) : 0.0f;
  b[1] = (lm < HID) ? (half ? b2[lm]           : W2[1 * HID + lm]) : 0.0f;
  v8f d2 = {};
  d2 = __builtin_amdgcn_wmma_f32_16x16x4_f32(false, a, false, b,
                                             (short)0, d2, false, false);
  __syncthreads();  // layer-2 A loads complete before h is rewritten
  #pragma unroll
  for (int v = 0; v < 8; ++v) {
    const int m = v + 8 * half;
    h[m * 16 + lm] = d2[v] > 0.0f ? d2[v] : 0.0f;
  }
  __syncthreads();

  // ---- layer 3: B[k][0] = W3[k] (k<3), B[3][0] = b3
  a[0] = h[lm * 16 + 2 * half];
  a[1] = half ? 1.0f : h[lm * 16 + 1];
  b[0] = (lm == 0) ? (half ? W3[2] : W3[0]) : 0.0f;
  b[1] = (lm == 0) ? (half ? b3[0] : W3[1]) : 0.0f;
  v8f d3 = {};
  d3 = __builtin_amdgcn_wmma_f32_16x16x4_f32(false, a, false, b,
                                             (short)0, d3, false, false);
  if (lm == 0) {  // column 0 of D lives in lanes 0 (M=v) and 16 (M=v+8)
    #pragma unroll
    for (int v = 0; v < 8; ++v) {
      const int m = v + 8 * half;
      out[cellBase + m] = 1.0f / (1.0f + __expf(-d3[v]));  // sigmoid
    }
  }
}

// ------------------------------------------------------------------ launch
extern "C" void kernel_launch(void* const* d_in, const int* in_sizes, int n_in,
                              void* d_out, int out_size, void* d_ws,
                              size_t ws_size, hipStream_t stream) {
  const float* X      = (const float*)d_in[0];
  const int*   rowIds = (const int*)d_in[1];
  const int*   colIds = (const int*)d_in[2];
  const float* W1     = (const float*)d_in[3];
  const float* b1     = (const float*)d_in[4];
  const float* W2     = (const float*)d_in[5];
  const float* b2     = (const float*)d_in[6];
  const float* W3     = (const float*)d_in[7];
  const float* b3     = (const float*)d_in[8];
  float* out = (float*)d_out;

  // workspace layout (4-byte elements)
  float* acc      = (float*)d_ws;                 // [4096] block sums
  int*   colStart = (int*)d_ws + RBK * CBK;       // [65]
  int*   rowCount = colStart + (CBK + 1);         // [64]
  int*   colCount = rowCount + RBK;               // [64]

  hipLaunchKernelGGL(k_zero_acc, dim3(16), dim3(256), 0, stream, acc);
  hipLaunchKernelGGL(k_counts, dim3(1), dim3(256), 0, stream,
                     rowIds, colIds, colStart, rowCount, colCount, out);
  hipLaunchKernelGGL(k_blocksum, dim3(CBK, NROWS / ROWS_PER_WG), dim3(256), 0,
                     stream, X, rowIds, colStart, acc);
  hipLaunchKernelGGL(k_mlp_wmma, dim3(RBK * CBK / 16), dim3(32), 0, stream,
                     acc, rowCount, colCount, W1, b1, W2, b2, W3, b3, out);
}